// HAN_21114059227346
// MI455X (gfx1250) — compile-verified
//
#include <hip/hip_runtime.h>
#include <hip/hip_bf16.h>
#include <math.h>

typedef __attribute__((ext_vector_type(2))) float v2f;
typedef __attribute__((ext_vector_type(8))) float v8f;

#define HIDF 64
#define NHEAD 4

// ---------------------------------------------------------------------------
// Generic fill
// ---------------------------------------------------------------------------
__global__ void fill_kernel(float* __restrict__ p, float v, int n) {
    int i = blockIdx.x * blockDim.x + threadIdx.x;
    if (i < n) p[i] = v;
}

// ---------------------------------------------------------------------------
// WMMA projection: Y[N,64] = X[N,K] @ W[K,64] + B[64]
// One wave computes a 16x16 tile via V_WMMA_F32_16X16X4_F32, K-loop step 4.
// Block = 128 threads (4 waves): wave w handles columns [16w, 16w+16).
// A layout (16x4 f32): lanes 0-15 hold M=lane, K={k,k+1} in vgpr0/1;
//                      lanes 16-31 hold M=lane-16, K={k+2,k+3}.
// B layout (4x16 f32): lanes 0-15 hold N=lane, K={k,k+1};
//                      lanes 16-31 hold N=lane-16, K={k+2,k+3}.
// C/D (16x16 f32): vgpr r: lanes 0-15 -> D[r][lane]; lanes 16-31 -> D[8+r][lane-16].
// ---------------------------------------------------------------------------
template <int K>
__global__ void proj_wmma_kernel(const float* __restrict__ X,
                                 const float* __restrict__ W,
                                 const float* __restrict__ B,
                                 float* __restrict__ Y) {
    const int wave = threadIdx.x >> 5;
    const int lane = threadIdx.x & 31;
    const int row0 = blockIdx.x << 4;
    const int col0 = wave << 4;
    const int m    = lane & 15;
    const int kh   = (lane >> 4) << 1;   // 0 or 2

    v8f acc = {};
#pragma unroll
    for (int k = 0; k < K; k += 4) {
        const float* ap = X + (size_t)(row0 + m) * K + (k + kh);
        v2f a;  a[0] = ap[0];      a[1] = ap[1];
        const float* bp = W + (size_t)(k + kh) * HIDF + (col0 + m);
        v2f b;  b[0] = bp[0];      b[1] = bp[HIDF];
        acc = __builtin_amdgcn_wmma_f32_16x16x4_f32(
                  false, a, false, b, (short)0, acc, false, false);
    }

    const int   col  = col0 + m;
    const int   rb   = row0 + ((lane >> 4) << 3);
    const float bias = B[col];
#pragma unroll
    for (int r = 0; r < 8; ++r)
        Y[(size_t)(rb + r) * HIDF + col] = acc[r] + bias;
}

// ---------------------------------------------------------------------------
// Per-node attention score: out[n,h] = sum_d xt[n, h*16+d] * avec[h*16+d]
// ---------------------------------------------------------------------------
__global__ void node_att_kernel(const float* __restrict__ xt,
                                const float* __restrict__ avec,
                                float* __restrict__ out, int N) {
    int id = blockIdx.x * blockDim.x + threadIdx.x;
    if (id >= N * NHEAD) return;
    int n = id >> 2, h = id & 3;
    const float* xr = xt + (size_t)n * HIDF + h * 16;
    const float* av = avec + h * 16;
    float s = 0.f;
#pragma unroll
    for (int d = 0; d < 16; ++d) s += xr[d] * av[d];
    out[id] = s;
}

// ---------------------------------------------------------------------------
// Edge alpha = leaky_relu(a_s[src]+a_d[dst], 0.2); segment max via CAS.
// ---------------------------------------------------------------------------
__device__ __forceinline__ void atomicMaxFloat(float* addr, float val) {
    unsigned int* a = (unsigned int*)addr;
    unsigned int old = *a;
    while (__uint_as_float(old) < val) {
        unsigned int assumed = old;
        old = atomicCAS(a, assumed, __float_as_uint(val));
        if (old == assumed) break;
    }
}

__global__ void edge_alpha_kernel(const int* __restrict__ src,
                                  const int* __restrict__ dst,
                                  const float* __restrict__ a_s,
                                  const float* __restrict__ a_d,
                                  float* __restrict__ alpha,
                                  float* __restrict__ segm, int E) {
    int id = blockIdx.x * blockDim.x + threadIdx.x;
    if (id >= E * NHEAD) return;
    int e = id >> 2, h = id & 3;
    int s = src[e], d = dst[e];
    float a = a_s[s * NHEAD + h] + a_d[d * NHEAD + h];
    a = (a > 0.f) ? a : 0.2f * a;
    alpha[id] = a;
    atomicMaxFloat(&segm[d * NHEAD + h], a);
}

// ---------------------------------------------------------------------------
// e = exp(alpha - segmax[dst]); segsum[dst] += e; alpha <- e (in place)
// ---------------------------------------------------------------------------
__global__ void edge_exp_kernel(const int* __restrict__ dst,
                                float* __restrict__ alpha,
                                const float* __restrict__ segm,
                                float* __restrict__ segs, int E) {
    int id = blockIdx.x * blockDim.x + threadIdx.x;
    if (id >= E * NHEAD) return;
    int e = id >> 2, h = id & 3;
    int d = dst[e];
    float ev = expf(alpha[id] - segm[d * NHEAD + h]);
    alpha[id] = ev;
    atomicAdd(&segs[d * NHEAD + h], ev);
}

// ---------------------------------------------------------------------------
// agg[dst, c] += xt_src[src, c] * e/(segsum[dst,h]+1e-16)
// ---------------------------------------------------------------------------
__global__ void edge_agg_kernel(const int* __restrict__ src,
                                const int* __restrict__ dst,
                                const float* __restrict__ xt_src,
                                const float* __restrict__ ealpha,
                                const float* __restrict__ segs,
                                float* __restrict__ agg, int E) {
    int id = blockIdx.x * blockDim.x + threadIdx.x;
    if (id >= E * HIDF) return;
    int e = id >> 6, c = id & 63, h = c >> 4;
    int s = src[e], d = dst[e];
    float w = ealpha[e * NHEAD + h] / (segs[d * NHEAD + h] + 1e-16f);
    atomicAdd(&agg[(size_t)d * HIDF + c], xt_src[(size_t)s * HIDF + c] * w);
}

__global__ void relu_kernel(float* __restrict__ p, int n) {
    int i = blockIdx.x * blockDim.x + threadIdx.x;
    if (i < n) { float v = p[i]; p[i] = v > 0.f ? v : 0.f; }
}

// ---------------------------------------------------------------------------
// Semantic attention partial: msum[c] += sum_n tanh((Z@Kw)[n,c] + Kb[c])
// Block of 256 = 64 cols x 4 node groups; 64 nodes per block.
// ---------------------------------------------------------------------------
__global__ void sem_partial_kernel(const float* __restrict__ Z,
                                   const float* __restrict__ Kw,
                                   const float* __restrict__ Kb,
                                   float* __restrict__ msum, int N) {
    __shared__ float red[256];
    int col = threadIdx.x & 63;
    int j   = threadIdx.x >> 6;
    int n0  = blockIdx.x * 64;
    float acc = 0.f;
    for (int i = 0; i < 16; ++i) {
        int n = n0 + j * 16 + i;
        if (n < N) {
            const float* zr = Z + (size_t)n * HIDF;
            float s = Kb[col];
#pragma unroll 8
            for (int c = 0; c < HIDF; ++c) s += zr[c] * Kw[c * HIDF + col];
            acc += tanhf(s);
        }
    }
    red[threadIdx.x] = acc;
    __syncthreads();
    if (j == 0) {
        float t = red[col] + red[col + 64] + red[col + 128] + red[col + 192];
        atomicAdd(&msum[col], t);
    }
}

__global__ void beta_kernel(const float* __restrict__ m0,
                            const float* __restrict__ m1,
                            const float* __restrict__ q,
                            float invN, float* __restrict__ beta) {
    float d0 = 0.f, d1 = 0.f;
    for (int c = 0; c < HIDF; ++c) {
        d0 += m0[c] * invN * q[c];
        d1 += m1[c] * invN * q[c];
    }
    float mx = fmaxf(d0, d1);
    float e0 = expf(d0 - mx), e1 = expf(d1 - mx);
    float s = e0 + e1;
    beta[0] = e0 / s;
    beta[1] = e1 / s;
}

__global__ void combine_kernel(const float* __restrict__ z0,
                               const float* __restrict__ z1,
                               const float* __restrict__ beta,
                               float* __restrict__ out, int n) {
    int i = blockIdx.x * blockDim.x + threadIdx.x;
    if (i < n) out[i] = beta[0] * z0[i] + beta[1] * z1[i];
}

// ---------------------------------------------------------------------------
// Final head: out[n,o] = X[n,:64] @ W[:,o] + b[o]
// ---------------------------------------------------------------------------
__global__ void final_lin_kernel(const float* __restrict__ X,
                                 const float* __restrict__ W,
                                 const float* __restrict__ B,
                                 float* __restrict__ out, int N) {
    int id = blockIdx.x * blockDim.x + threadIdx.x;
    if (id >= N * 16) return;
    int n = id >> 4, o = id & 15;
    const float* xr = X + (size_t)n * HIDF;
    float s = B[o];
#pragma unroll 8
    for (int c = 0; c < HIDF; ++c) s += xr[c] * W[c * 16 + o];
    out[id] = s;
}

// ---------------------------------------------------------------------------
extern "C" void kernel_launch(void* const* d_in, const int* in_sizes, int n_in,
                              void* d_out, int out_size, void* d_ws, size_t ws_size,
                              hipStream_t stream) {
    const int NU = 50000, NI = 50000, E = 800000;

    const float* x_user = (const float*)d_in[0];
    const float* x_item = (const float*)d_in[1];
    const int* ei_u2i = (const int*)d_in[2];
    const int* ei_i2u = (const int*)d_in[3];
    const int* ei_u2u = (const int*)d_in[4];
    // params in setup_inputs insertion order:
    // per layer: proj_user_w, proj_user_b, proj_item_w, proj_item_b,
    //            asrc_u2i, adst_u2i, asrc_i2u, adst_i2u, asrc_u2u, adst_u2u,
    //            k_w, k_b, q   (13 entries); then lin_w, lin_b
    const float* P[28];
    for (int i = 0; i < 28; ++i) P[i] = (const float*)d_in[5 + i];

    // Workspace carve-up (floats)
    float* ws = (float*)d_ws;
    size_t off = 0;
    auto carve = [&](size_t n) { float* p = ws + off; off += (n + 63) & ~(size_t)63; return p; };
    float* xt_user = carve((size_t)NU * HIDF);
    float* xt_item = carve((size_t)NI * HIDF);
    float* xu_next = carve((size_t)NU * HIDF);
    float* a_s     = carve((size_t)NU * NHEAD);
    float* a_d     = carve((size_t)NU * NHEAD);
    float* alpha   = carve((size_t)E * NHEAD);
    float* segm    = carve((size_t)NU * NHEAD);
    float* segs    = carve((size_t)NU * NHEAD);
    float* agg_u2i = carve((size_t)NI * HIDF);
    float* agg_i2u = carve((size_t)NU * HIDF);
    float* agg_u2u = carve((size_t)NU * HIDF);
    float* msum0   = carve(64);
    float* msum1   = carve(64);
    float* betaP   = carve(2);

    const int T = 256;
    auto blocks = [](long long n, int t) { return (int)((n + t - 1) / t); };

    auto fill = [&](float* p, float v, int n) {
        fill_kernel<<<blocks(n, T), T, 0, stream>>>(p, v, n);
    };

    // One complete edge-type pass (attention softmax + weighted aggregation + relu)
    auto run_edges = [&](const int* ei, const float* xts, int Ns,
                         const float* xtd, int Nd,
                         const float* asrc, const float* adst, float* agg) {
        const int* srcI = ei;
        const int* dstI = ei + E;
        fill(segm, -1e30f, Nd * NHEAD);
        fill(segs, 0.f, Nd * NHEAD);
        fill(agg, 0.f, Nd * HIDF);
        node_att_kernel<<<blocks((long long)Ns * NHEAD, T), T, 0, stream>>>(xts, asrc, a_s, Ns);
        node_att_kernel<<<blocks((long long)Nd * NHEAD, T), T, 0, stream>>>(xtd, adst, a_d, Nd);
        edge_alpha_kernel<<<blocks((long long)E * NHEAD, T), T, 0, stream>>>(
            srcI, dstI, a_s, a_d, alpha, segm, E);
        edge_exp_kernel<<<blocks((long long)E * NHEAD, T), T, 0, stream>>>(
            dstI, alpha, segm, segs, E);
        edge_agg_kernel<<<blocks((long long)E * HIDF, T), T, 0, stream>>>(
            srcI, dstI, xts, alpha, segs, agg, E);
        relu_kernel<<<blocks((long long)Nd * HIDF, T), T, 0, stream>>>(agg, Nd * HIDF);
    };

    // Semantic (metapath) attention for the user node type: z0 = i2u agg, z1 = u2u agg
    auto run_sem = [&](const float* z0, const float* z1,
                       const float* kw, const float* kb, const float* q, float* dst) {
        fill(msum0, 0.f, 64);
        fill(msum1, 0.f, 64);
        int sb = blocks(NU, 64);
        sem_partial_kernel<<<sb, 256, 0, stream>>>(z0, kw, kb, msum0, NU);
        sem_partial_kernel<<<sb, 256, 0, stream>>>(z1, kw, kb, msum1, NU);
        beta_kernel<<<1, 1, 0, stream>>>(msum0, msum1, q, 1.f / (float)NU, betaP);
        combine_kernel<<<blocks((long long)NU * HIDF, T), T, 0, stream>>>(
            z0, z1, betaP, dst, NU * HIDF);
    };

    // ---------------- Layer 0 (input dim 128) ----------------
    proj_wmma_kernel<128><<<NU / 16, 128, 0, stream>>>(x_user, P[0], P[1], xt_user);
    proj_wmma_kernel<128><<<NI / 16, 128, 0, stream>>>(x_item, P[2], P[3], xt_item);
    run_edges(ei_u2i, xt_user, NU, xt_item, NI, P[4], P[5], agg_u2i);   // dst=item
    run_edges(ei_i2u, xt_item, NI, xt_user, NU, P[6], P[7], agg_i2u);   // dst=user (z0)
    run_edges(ei_u2u, xt_user, NU, xt_user, NU, P[8], P[9], agg_u2u);   // dst=user (z1)
    run_sem(agg_i2u, agg_u2u, P[10], P[11], P[12], xu_next);            // new x_user
    // new x_item = agg_u2i (item semantic attn has P=1 -> beta=1, identity)

    // ---------------- Layer 1 (input dim 64) ----------------
    // u2i (dst=item) and the item update are dead code: output reads only x_user.
    proj_wmma_kernel<64><<<NU / 16, 128, 0, stream>>>(xu_next, P[13], P[14], xt_user);
    proj_wmma_kernel<64><<<NI / 16, 128, 0, stream>>>(agg_u2i, P[15], P[16], xt_item);
    run_edges(ei_i2u, xt_item, NI, xt_user, NU, P[19], P[20], agg_i2u); // z0
    run_edges(ei_u2u, xt_user, NU, xt_user, NU, P[21], P[22], agg_u2u); // z1
    run_sem(agg_i2u, agg_u2u, P[23], P[24], P[25], xu_next);            // final x_user

    // ---------------- Output head ----------------
    final_lin_kernel<<<blocks((long long)NU * 16, T), T, 0, stream>>>(
        xu_next, P[26], P[27], (float*)d_out, NU);
}